// PersonalizedFeaturesLayer_3212635538190
// MI455X (gfx1250) — compile-verified
//
#include <hip/hip_runtime.h>
#include <hip/hip_bf16.h>

typedef float v8f __attribute__((ext_vector_type(8)));
typedef float v2f __attribute__((ext_vector_type(2)));

#define DIM        128
#define HLEN       200
#define HPAD       208          // 13 * 16 row tiles
#define HSTRIDE    132          // 128 + 4 pad -> conflict-free column reads
#define AHID       64
#define NTHREADS   256

#define LDS_AS __attribute__((address_space(3)))
static __device__ __forceinline__ unsigned lds_offset_of(const void* p) {
    // generic -> LDS addrspacecast strips the aperture, leaving the byte offset
    return (unsigned)(uintptr_t)(LDS_AS const void*)p;
}

__global__ __launch_bounds__(NTHREADS)
void personalized_features_kernel(const int*   __restrict__ user_ids,
                                  const int*   __restrict__ item_ids,
                                  const int*   __restrict__ user_history,
                                  const float* __restrict__ user_table,
                                  const float* __restrict__ item_table,
                                  const float* __restrict__ fi,      // [DIM, DIM]
                                  const float* __restrict__ w1,      // [2*DIM, AHID]
                                  const float* __restrict__ b1,      // [AHID]
                                  const float* __restrict__ w2,      // [AHID]
                                  const float* __restrict__ b2,      // [1]
                                  float* __restrict__ out_user,
                                  float* __restrict__ out_item,
                                  float* __restrict__ out_inter)
{
    extern __shared__ float smem[];
    float* s_hist  = smem;                          // HPAD * HSTRIDE
    float* s_u     = s_hist  + HPAD * HSTRIDE;      // DIM
    float* s_item  = s_u     + DIM;                 // DIM
    float* s_logit = s_item  + DIM;                 // HPAD
    float* s_attn  = s_logit + HPAD;                // HPAD
    float* s_ubias = s_attn  + HPAD;                // AHID: b1[n] + dot(u, w1_user[:,n])
    float* s_red   = s_ubias + AHID;                // 8

    const int b    = blockIdx.x;
    const int tid  = threadIdx.x;
    const int lane = tid & 31;
    const int wave = tid >> 5;
    const int half = lane >> 4;                     // 0: lanes 0-15, 1: lanes 16-31
    const int l16  = lane & 15;

    // ---------------- stage 0: stage embeddings + history into LDS ----------------
    const int uid = user_ids[b];
    const int iid = item_ids[b];
    if (tid < DIM)      s_u[tid]          = user_table[(size_t)uid * DIM + tid];
    else                s_item[tid - DIM] = item_table[(size_t)iid * DIM + (tid - DIM)];

    for (int i = tid; i < HPAD; i += NTHREADS) s_logit[i] = 0.0f;
    if (tid < AHID) s_ubias[tid] = b1[tid];
    // zero the 8 pad rows (only first DIM columns are ever read)
    for (int i = tid; i < 8 * DIM; i += NTHREADS) {
        int r = HLEN + (i >> 7);
        int c = i & (DIM - 1);
        s_hist[r * HSTRIDE + c] = 0.0f;
    }

    // async gather: 200 item rows straight into LDS (no VGPR round trip),
    // 16B per lane per issue, tracked by ASYNCcnt
    {
        const float4* it4 = (const float4*)item_table;
        for (int idx = tid; idx < HLEN * (DIM / 4); idx += NTHREADS) {
            int row = idx >> 5;                     // DIM/4 == 32 quads per row
            int q   = idx & 31;
            int id  = user_history[b * HLEN + row];
            const float4* src = it4 + (size_t)id * (DIM / 4) + q;
            unsigned loff = lds_offset_of(&s_hist[row * HSTRIDE + q * 4]);
            asm volatile("global_load_async_to_lds_b128 %0, %1, off"
                         :: "v"(loff), "v"(src) : "memory");
        }
    }
    __syncthreads();                                // s_u / s_ubias init visible

    // ---------------- stage 0.5: fold user half into the bias ----------------
    // s_ubias[n] += dot(u, w1[0:128, n])  (4 partial threads per n)
    {
        const int n    = tid & (AHID - 1);
        const int part = tid >> 6;                  // 0..3 -> k chunks of 32
        float acc = 0.0f;
        #pragma unroll 8
        for (int k = part * 32; k < part * 32 + 32; ++k)
            acc = fmaf(s_u[k], w1[k * AHID + n], acc);
        atomicAdd(&s_ubias[n], acc);
    }
    asm volatile("s_wait_asynccnt 0x0" ::: "memory");   // this wave's async gathers done
    __syncthreads();                                    // all waves' gathers + ubias done

    // ---------------- stage 1: history-half GEMM via V_WMMA_F32_16X16X4_F32 ----------
    // wave -> fixed N-tile; B fragments (w1 rows 128..255) held in 64 VGPRs
    {
        const int nt    = wave & 3;
        const int mt0   = (wave < 4) ? 0 : 7;
        const int mtEnd = (wave < 4) ? 7 : 13;
        const int n     = nt * 16 + l16;
        const float w2n     = w2[n];
        const float ubias_n = s_ubias[n];

        v2f breg[32];
        #pragma unroll
        for (int j = 0; j < 32; ++j) {
            const int ko = DIM + j * 4 + half * 2;
            breg[j].x = w1[ko * AHID + n];
            breg[j].y = w1[(ko + 1) * AHID + n];
        }

        for (int mt = mt0; mt < mtEnd; ++mt) {
            const int arow  = mt * 16 + l16;
            const float* ap = &s_hist[arow * HSTRIDE + half * 2];

            v8f c;
            #pragma unroll
            for (int i = 0; i < 8; ++i) c[i] = ubias_n;   // bias + user-half, all rows

            #pragma unroll
            for (int j = 0; j < 32; ++j) {
                v2f a = *(const v2f*)(ap + j * 4);        // ds_load_b64, conflict-free
                c = __builtin_amdgcn_wmma_f32_16x16x4_f32(false, a, false, breg[j],
                                                          (short)0, c, false, false);
            }

            // fused epilogue: relu * w2[n], reduce 16 N-lanes per row, logit += (ds_add_f32)
            #pragma unroll
            for (int i = 0; i < 8; ++i) {
                float v = c[i];
                v = v > 0.0f ? v : 0.0f;
                v *= w2n;
                v += __shfl_xor(v, 8, 32);
                v += __shfl_xor(v, 4, 32);
                v += __shfl_xor(v, 2, 32);
                v += __shfl_xor(v, 1, 32);
                if (l16 == 0)
                    atomicAdd(&s_logit[mt * 16 + half * 8 + i], v);
            }
        }
    }
    __syncthreads();

    // ---------------- stage 2: sigmoid ----------------
    if (tid < HLEN) {
        float x = s_logit[tid] + b2[0];
        s_attn[tid] = 1.0f / (1.0f + expf(-x));
    }
    __syncthreads();

    // ---------------- stage 3: pooling (waves 0-3) | bilinear (waves 4-7) ----------------
    if (wave < 4) {
        const int d = wave * 32 + lane;             // 0..127
        float acc = 0.0f;
        for (int h = 0; h < HLEN; ++h)
            acc = fmaf(s_attn[h], s_hist[h * HSTRIDE + d], acc);
        out_user[(size_t)b * DIM + d] = s_u[d] + acc;
        out_item[(size_t)b * DIM + d] = s_item[d];
    } else {
        const int j = (wave - 4) * 32 + lane;       // 0..127
        float acc = 0.0f;
        for (int k = 0; k < DIM; ++k)
            acc = fmaf(s_u[k], fi[k * DIM + j], acc);   // coalesced column reads
        float p = acc * s_item[j];
        p += __shfl_xor(p, 16, 32);
        p += __shfl_xor(p, 8, 32);
        p += __shfl_xor(p, 4, 32);
        p += __shfl_xor(p, 2, 32);
        p += __shfl_xor(p, 1, 32);
        if (lane == 0) s_red[wave - 4] = p;
    }
    __syncthreads();

    if (tid == 0)
        out_inter[b] = s_red[0] + s_red[1] + s_red[2] + s_red[3];
}

extern "C" void kernel_launch(void* const* d_in, const int* in_sizes, int n_in,
                              void* d_out, int out_size, void* d_ws, size_t ws_size,
                              hipStream_t stream) {
    const int* user_ids     = (const int*)d_in[0];
    const int* item_ids     = (const int*)d_in[1];
    const int* user_history = (const int*)d_in[2];
    const float* user_table = (const float*)d_in[3];
    const float* item_table = (const float*)d_in[4];
    const float* fi         = (const float*)d_in[5];
    const float* w1         = (const float*)d_in[6];
    const float* b1         = (const float*)d_in[7];
    const float* w2         = (const float*)d_in[8];
    const float* b2         = (const float*)d_in[9];

    const int B = in_sizes[0];                      // 8192
    float* out      = (float*)d_out;
    float* out_user = out;                          // [B, DIM]
    float* out_item = out + (size_t)B * DIM;        // [B, DIM]
    float* out_int  = out + (size_t)2 * B * DIM;    // [B, 1]

    const size_t shmem = (size_t)(HPAD * HSTRIDE + DIM + DIM + HPAD + HPAD + AHID + 8)
                         * sizeof(float);

    personalized_features_kernel<<<B, NTHREADS, shmem, stream>>>(
        user_ids, item_ids, user_history, user_table, item_table,
        fi, w1, b1, w2, b2, out_user, out_item, out_int);
}